// EdgeScore_function_67001489817643
// MI455X (gfx1250) — compile-verified
//
#include <hip/hip_runtime.h>

#define NDIM 1024
#define DDIM 1024
#define HDIM 100
#define HT   7            // ceil(100/16) h-tiles
#define HPAD 112          // HT*16, workspace row stride
#define LSTR 101          // LDS row stride (bank-conflict-free for 8 j-groups)

typedef __attribute__((ext_vector_type(2))) float v2f;
typedef __attribute__((ext_vector_type(8))) float v8f;

#if __has_builtin(__builtin_amdgcn_tanhf)
__device__ __forceinline__ float fast_tanh(float x) { return __builtin_amdgcn_tanhf(x); }
#else
__device__ __forceinline__ float fast_tanh(float x) { return tanhf(x); }
#endif

// ---------------------------------------------------------------------------
// Kernel 1: one wave per (16-row tile, half) computes a 16 x 112 strip of
//   half==0:  L[n,h] = sum_d E[n,d] * W1[h, d]
//   half==1:  R[n,h] = sum_d E[n,d] * W1[h, D+d] + b1[h]
// using V_WMMA_F32_16X16X4_F32 accumulation (full f32 precision).
// Fragment layouts per CDNA5 ISA 7.12.2:
//   A (16x4):  lanes 0-15 hold M=lane, VGPR0/1 = K0/K1; lanes 16-31 = K2/K3
//   B (4x16):  lanes 0-15 hold N=lane, VGPR0/1 = K0/K1; lanes 16-31 = K2/K3
//   D (16x16): VGPR r: lanes 0-15 -> M=r, N=lane; lanes 16-31 -> M=r+8
// Lanes whose h >= 100 (last tile) load a CLAMPED W1 row (99) instead of
// branching: their results land only in padding columns 100..111 of the
// workspace, which kernel 2 never reads. Keeps the k-loop branch-free.
// ---------------------------------------------------------------------------
__global__ __launch_bounds__(32)
void edgescore_gemm_lr(const float* __restrict__ E,
                       const float* __restrict__ W1,
                       const float* __restrict__ b1,
                       float* __restrict__ LR)   // [2][NDIM][HPAD]
{
    const int lane     = threadIdx.x;      // 0..31
    const int rowTile  = blockIdx.x;       // 0..63
    const int half     = blockIdx.y;       // 0 = left(A), 1 = right(B)
    const int halfLane = lane & 15;
    const int hiHalf   = lane >> 4;        // 0: K0/K1, 1: K2/K3

    // A-fragment source row (E row for this lane) + K-pair offset
    const int n = rowTile * 16 + halfLane;
    const float* __restrict__ Erow = E + (size_t)n * DDIM + (hiHalf ? 2 : 0);

    // B-fragment row pointers (one per h-tile), clamped for padding lanes
    const int wOff = half * DDIM;          // which half of W1's 2D columns
    const float* Wrow[HT];
#pragma unroll
    for (int t = 0; t < HT; ++t) {
        int h = t * 16 + halfLane;
        if (h > HDIM - 1) h = HDIM - 1;    // clamp: garbage lands in padding only
        Wrow[t] = W1 + (size_t)h * (2 * DDIM) + wOff + (hiHalf ? 2 : 0);
    }

    v8f acc[HT];
#pragma unroll
    for (int t = 0; t < HT; ++t) acc[t] = (v8f)0.0f;

    for (int k = 0; k < DDIM; k += 4) {
        const v2f a = *(const v2f*)(Erow + k);   // E[n, k + 2*hiHalf .. +1]
#pragma unroll
        for (int t = 0; t < HT; ++t) {
            const v2f b = *(const v2f*)(Wrow[t] + k);
            acc[t] = __builtin_amdgcn_wmma_f32_16x16x4_f32(
                         false, a, false, b, (short)0, acc[t], false, false);
        }
    }

    // Epilogue: fold b1 into the right half; store strip (h>=100 is padding).
    float* __restrict__ out = LR + (size_t)half * NDIM * HPAD;
#pragma unroll
    for (int t = 0; t < HT; ++t) {
        const int h = t * 16 + halfLane;                  // N index (same for all 8 regs)
        const float bv = (half == 1 && h < HDIM) ? b1[h] : 0.0f;
#pragma unroll
        for (int r = 0; r < 8; ++r) {
            const int m    = r + hiHalf * 8;              // M index within tile
            const int nrow = rowTile * 16 + m;
            out[(size_t)nrow * HPAD + h] = acc[t][r] + bv;
        }
    }
}

// ---------------------------------------------------------------------------
// Kernel 2: scores[i,j] = b2 + sum_h W2[h] * tanh(L[i,h] + R[j,h])
// 32x32 output tile per 256-thread block; each thread: 1 i x 4 j.
// L/R/W2 staged in LDS (stride 101 => conflict-free R reads, broadcast L/W2).
// Dominant op: native v_tanh_f32 on the TRANS pipe (co-executes with VALU).
// ---------------------------------------------------------------------------
__global__ __launch_bounds__(256)
void edgescore_scores(const float* __restrict__ LR,
                      const float* __restrict__ W2,
                      const float* __restrict__ b2,
                      float* __restrict__ out)
{
    __shared__ float sL[32 * LSTR];
    __shared__ float sR[32 * LSTR];
    __shared__ float sW[HDIM];

    const int tid = threadIdx.x;
    const int ib  = blockIdx.x * 32;
    const int jb  = blockIdx.y * 32;

    const float* __restrict__ Lg = LR;                          // [N][HPAD]
    const float* __restrict__ Rg = LR + (size_t)NDIM * HPAD;    // [N][HPAD]

    for (int idx = tid; idx < 32 * HDIM; idx += 256) {
        const int r = idx / HDIM, c = idx % HDIM;
        sL[r * LSTR + c] = Lg[(size_t)(ib + r) * HPAD + c];
        sR[r * LSTR + c] = Rg[(size_t)(jb + r) * HPAD + c];
    }
    if (tid < HDIM) sW[tid] = W2[tid];
    __syncthreads();

    const int iloc = tid >> 3;   // 0..31 (4 distinct i per wave -> LDS broadcast)
    const int jg   = tid & 7;    // 0..7  (8 distinct j-groups -> 8 distinct banks)

    const float* __restrict__ lrow  = sL + iloc * LSTR;
    const float* __restrict__ rbase = sR + (jg * 4) * LSTR;

    float a0 = 0.f, a1 = 0.f, a2 = 0.f, a3 = 0.f;
#pragma unroll 4
    for (int h = 0; h < HDIM; ++h) {
        const float w  = sW[h];
        const float lv = lrow[h];
        const float t0 = fast_tanh(lv + rbase[h]);
        const float t1 = fast_tanh(lv + rbase[LSTR + h]);
        const float t2 = fast_tanh(lv + rbase[2 * LSTR + h]);
        const float t3 = fast_tanh(lv + rbase[3 * LSTR + h]);
        a0 = fmaf(w, t0, a0);
        a1 = fmaf(w, t1, a1);
        a2 = fmaf(w, t2, a2);
        a3 = fmaf(w, t3, a3);
    }

    const float bb = b2[0];
    float4 res = make_float4(a0 + bb, a1 + bb, a2 + bb, a3 + bb);
    *(float4*)(out + (size_t)(ib + iloc) * NDIM + jb + jg * 4) = res;
}

// ---------------------------------------------------------------------------
extern "C" void kernel_launch(void* const* d_in, const int* in_sizes, int n_in,
                              void* d_out, int out_size, void* d_ws, size_t ws_size,
                              hipStream_t stream) {
    const float* E  = (const float*)d_in[0];   // (N, D)
    const float* W1 = (const float*)d_in[1];   // (H, 2D)
    const float* b1 = (const float*)d_in[2];   // (H,)
    const float* W2 = (const float*)d_in[3];   // (1, H)
    const float* b2 = (const float*)d_in[4];   // (1,)
    float* out = (float*)d_out;                // (N, N)
    float* LR  = (float*)d_ws;                 // 2 * N * HPAD floats = 896 KiB

    dim3 g1(NDIM / 16, 2);
    edgescore_gemm_lr<<<g1, 32, 0, stream>>>(E, W1, b1, LR);

    dim3 g2(NDIM / 32, NDIM / 32);
    edgescore_scores<<<g2, 256, 0, stream>>>(LR, W2, b2, out);
}